// MeanPooling_4329327034746
// MI455X (gfx1250) — compile-verified
//
#include <hip/hip_runtime.h>

// Problem constants (from the reference): B=16, T=2048, D=1024, fp32.
#define B_N 16
#define T_N 2048
#define D_N 1024

#define TCHUNK 64                  // T-elements per wave (32 chunks -> 8192 waves, ~8/SIMD)
#define NTCHUNK (T_N / TCHUNK)     // 32
#define DTILES 4                   // 16-wide d-tiles per wave -> 64 d's
#define DGROUPS (D_N / (16 * DTILES))  // 16

typedef __attribute__((ext_vector_type(2))) float v2f;
typedef __attribute__((ext_vector_type(8))) float v8f;

// Zero-init the output accumulator (d_out is poisoned by the harness; this
// runs every launch so graph replays stay deterministic).
__global__ void MeanPooling_zero_kernel(float* __restrict__ out) {
    int i = blockIdx.x * blockDim.x + threadIdx.x;
    if (i < B_N * D_N) out[i] = 0.0f;
}

// One wave (32 lanes) per block. Computes, for its (b, 64-d slice, T-chunk):
//   out[b, d] += sum_{t in chunk, t < len} xs[b,t,d] * (1/len)
// using V_WMMA_F32_16X16X4_F32:
//   A (16x4)  : mask/len, identical in all 16 rows
//   B (4x16)  : xs tile, K = t, N = d
//   D row M=0 : the 16 masked-scaled partial sums (VGPR0, lanes 0-15)
__global__ __launch_bounds__(32)
void MeanPooling_wmma_kernel(const float* __restrict__ xs,
                             const int* __restrict__ xs_len,
                             float* __restrict__ out) {
    const int lane   = threadIdx.x & 31;
    const int tchunk = blockIdx.x;
    const int dgroup = blockIdx.y;
    const int b      = blockIdx.z;

    const int n     = lane & 15;        // N (d) index within tile
    const int khalf = (lane >> 4) << 1; // K offset: lanes 0-15 -> K0/K1, 16-31 -> K2/K3

    const int len = xs_len[b];
    const int tbase = tchunk * TCHUNK;
    if (tbase >= len) return;           // fully masked chunk: contributes nothing

    const float inv_len = 1.0f / (float)len;
    const int d0 = dgroup * (16 * DTILES);
    const float* __restrict__ base = xs + (size_t)b * T_N * D_N + d0 + n;

    v8f acc0 = {}; v8f acc1 = {}; v8f acc2 = {}; v8f acc3 = {};

    // Clamp trip count to len (rounded up to the K=4 granule): masked-out
    // remainder lanes get A=0, fully-masked steps are skipped entirely.
    int tstop = tbase + TCHUNK;
    int lstop = (len + 3) & ~3;
    if (lstop < tstop) tstop = lstop;

#pragma unroll 2
    for (int t0 = tbase; t0 < tstop; t0 += 4) {
        const int ta = t0 + khalf;
        v2f A;
        A.x = (ta     < len) ? inv_len : 0.0f;
        A.y = (ta + 1 < len) ? inv_len : 0.0f;

        const float* p = base + (size_t)ta * D_N;  // xs[b, ta, d0+n]

        v2f Bv0; Bv0.x = p[0];      Bv0.y = p[0      + D_N];
        v2f Bv1; Bv1.x = p[16];     Bv1.y = p[16     + D_N];
        v2f Bv2; Bv2.x = p[32];     Bv2.y = p[32     + D_N];
        v2f Bv3; Bv3.x = p[48];     Bv3.y = p[48     + D_N];

        acc0 = __builtin_amdgcn_wmma_f32_16x16x4_f32(false, A, false, Bv0,
                                                     (short)0, acc0, false, false);
        acc1 = __builtin_amdgcn_wmma_f32_16x16x4_f32(false, A, false, Bv1,
                                                     (short)0, acc1, false, false);
        acc2 = __builtin_amdgcn_wmma_f32_16x16x4_f32(false, A, false, Bv2,
                                                     (short)0, acc2, false, false);
        acc3 = __builtin_amdgcn_wmma_f32_16x16x4_f32(false, A, false, Bv3,
                                                     (short)0, acc3, false, false);
    }

    // D-matrix row M=0 lives in accumulator element 0, lanes 0-15.
    if (lane < 16) {
        float* o = out + (size_t)b * D_N + d0 + n;
        atomicAdd(o + 0,  acc0[0]);
        atomicAdd(o + 16, acc1[0]);
        atomicAdd(o + 32, acc2[0]);
        atomicAdd(o + 48, acc3[0]);
    }
}

extern "C" void kernel_launch(void* const* d_in, const int* in_sizes, int n_in,
                              void* d_out, int out_size, void* d_ws, size_t ws_size,
                              hipStream_t stream) {
    const float* xs     = (const float*)d_in[0];
    const int*   xs_len = (const int*)d_in[1];
    float*       out    = (float*)d_out;

    MeanPooling_zero_kernel<<<dim3((B_N * D_N + 255) / 256), dim3(256), 0, stream>>>(out);

    dim3 grid(NTCHUNK, DGROUPS, B_N);
    MeanPooling_wmma_kernel<<<grid, dim3(32), 0, stream>>>(xs, xs_len, out);
}